// Equiformer_MD17_DeNS_VAE_67723044323874
// MI455X (gfx1250) — compile-verified
//
#include <hip/hip_runtime.h>
#include <math.h>

#define NNODES 20000
#define NEDGES 320000
#define SDIM 256
#define FDIM 960
#define NBASIS 32
#define CUTR 5.0f

typedef float v2f __attribute__((ext_vector_type(2)));
typedef float v8f __attribute__((ext_vector_type(8)));

__device__ __forceinline__ float siluf(float x) { return x / (1.0f + expf(-x)); }

// order-preserving float<->uint mapping for atomicMax-based segment max
__device__ __forceinline__ unsigned f2ord(float f) {
  unsigned b = __float_as_uint(f);
  return (b & 0x80000000u) ? ~b : (b | 0x80000000u);
}
__device__ __forceinline__ float ord2f(unsigned u) {
  return (u & 0x80000000u) ? __uint_as_float(u ^ 0x80000000u) : __uint_as_float(~u);
}

// ---------------------------------------------------------------------------
// WMMA fp32 GEMM: C[M x 256] = act(A[M x 256] @ B[256 x 256] + bias)
// A row-major (lda), B row-major (ldb=256), C row-major (ldc=256).
// Tile: 128(M) x 64(N), K staged in chunks of 32, v_wmma_f32_16x16x4_f32.
// ---------------------------------------------------------------------------
#define BM 128
#define BN 64
#define KT 32
#define TS 34   // LDS row stride (even -> 8B aligned b64, conflict-free)

__global__ __launch_bounds__(256) void gemm256(
    const float* __restrict__ A, int lda, int M,
    const float* __restrict__ B, int ldb,
    const float* __restrict__ bias,
    float* __restrict__ C, int ldc, int act)
{
  __shared__ float As[BM * TS];   // [m][k]
  __shared__ float Bs[BN * TS];   // [n][k]  (transposed stage)
  const int tid  = threadIdx.x;
  const int lane = tid & 31;
  const int wv   = tid >> 5;         // 8 waves: wave wv owns rows [wv*16, wv*16+16)
  const int half = lane >> 4;
  const int lrow = lane & 15;
  const int m0 = blockIdx.x * BM;
  const int n0 = blockIdx.y * BN;

  v8f acc[4];
#pragma unroll
  for (int s = 0; s < 4; ++s)
#pragma unroll
    for (int r = 0; r < 8; ++r) acc[s][r] = 0.0f;

  for (int kt = 0; kt < SDIM / KT; ++kt) {
    __syncthreads();
    // stage A tile 128x32 (zero-padded past M)
#pragma unroll
    for (int it = 0; it < 4; ++it) {
      int r = it * 32 + (tid >> 3);
      int c = (tid & 7) << 2;
      int gm = m0 + r;
      float4 v = make_float4(0.f, 0.f, 0.f, 0.f);
      if (gm < M) v = *(const float4*)(A + (size_t)gm * lda + kt * KT + c);
      As[r * TS + c + 0] = v.x;
      As[r * TS + c + 1] = v.y;
      As[r * TS + c + 2] = v.z;
      As[r * TS + c + 3] = v.w;
    }
    // stage B tile 32x64 -> Bs[n][k]
    {
      int n = tid & 63;
      int kk0 = tid >> 6;
#pragma unroll
      for (int kk = 0; kk < KT; kk += 4)
        Bs[n * TS + kk + kk0] = B[(size_t)(kt * KT + kk + kk0) * ldb + n0 + n];
    }
    __syncthreads();
#pragma unroll
    for (int ks = 0; ks < KT / 4; ++ks) {
      int k = ks * 4 + 2 * half;  // low lanes K0/K1, high lanes K2/K3 (ISA layout)
      v2f a = *(const v2f*)&As[(wv * 16 + lrow) * TS + k];
#pragma unroll
      for (int s = 0; s < 4; ++s) {
        v2f b = *(const v2f*)&Bs[(s * 16 + lrow) * TS + k];
        acc[s] = __builtin_amdgcn_wmma_f32_16x16x4_f32(
            false, a, false, b, (short)0, acc[s], false, false);
      }
    }
  }

  // epilogue: D vgpr r -> row (r + 8*half), col = lrow of subtile
#pragma unroll
  for (int s = 0; s < 4; ++s) {
    int col = n0 + s * 16 + lrow;
    float bv = bias ? bias[col] : 0.0f;
#pragma unroll
    for (int r = 0; r < 8; ++r) {
      int row = m0 + wv * 16 + half * 8 + r;
      if (row < M) {
        float v = acc[s][r] + bv;
        if (act) v = siluf(v);
        C[(size_t)row * ldc + col] = v;
      }
    }
  }
}

// ---------------------------------------------------------------------------
// ai = exp(Hai @ W_ai2 + b_ai2), ci = exp(Hci @ W_ci2 + b_ci2); bsum init = ci
// one wave per node, lane handles channels lane+32i
// ---------------------------------------------------------------------------
__global__ __launch_bounds__(256) void aici_kernel(
    const float* __restrict__ Hai, const float* __restrict__ Hci,
    const float* __restrict__ Wai2, const float* __restrict__ bai2,
    const float* __restrict__ Wci2, const float* __restrict__ bci2,
    float* __restrict__ out_ai, float* __restrict__ bsum)
{
  int gw = (blockIdx.x * blockDim.x + threadIdx.x) >> 5;
  int lane = threadIdx.x & 31;
  if (gw >= NNODES) return;
  float sa = 0.f, sc = 0.f;
#pragma unroll
  for (int i = 0; i < 8; ++i) {
    int c = lane + 32 * i;
    sa += Hai[(size_t)gw * SDIM + c] * Wai2[c];
    sc += Hci[(size_t)gw * SDIM + c] * Wci2[c];
  }
#pragma unroll
  for (int m = 16; m >= 1; m >>= 1) {
    sa += __shfl_xor(sa, m, 32);
    sc += __shfl_xor(sc, m, 32);
  }
  if (lane == 0) {
    out_ai[gw] = expf(sa + bai2[0]);
    bsum[gw] = expf(sc + bci2[0]);
  }
}

// bvec[c] = b_b1[c] + sum_k b_ed[k] * W_b1[(512+k)*256 + c]
__global__ void bvec_kernel(const float* __restrict__ bed,
                            const float* __restrict__ Wb1,
                            const float* __restrict__ bb1,
                            float* __restrict__ bvec)
{
  int c = threadIdx.x;
  float s = bb1[c];
  for (int k = 0; k < SDIM; ++k) s += bed[k] * Wb1[(size_t)(2 * SDIM + k) * SDIM + c];
  bvec[c] = s;
}

__global__ void init_kernel(float* __restrict__ out, unsigned* __restrict__ bmaxk)
{
  int i = blockIdx.x * blockDim.x + threadIdx.x;
  if (i < 9 * NNODES) out[i] = 0.0f;                   // sigma accumulators
  if (i < NNODES) {
    bmaxk[i] = 0u;                                      // ordered key of -inf
    out[10 * NNODES + i] = 0.0f;                        // sum_gamma
  }
}

// ---------------------------------------------------------------------------
// Pass E1: one wave per edge. h = U[src] + V[dst] + bvec + rbf(d) @ Wc ;
// bij = silu(h) . W_b2 + b_b2 ; atomic segment-max into bmax keys.
// ---------------------------------------------------------------------------
__global__ __launch_bounds__(256) void edge_bij_kernel(
    const float* __restrict__ U, const float* __restrict__ V,
    const int* __restrict__ ei, const float* __restrict__ pos,
    const float* __restrict__ Wc, const float* __restrict__ bvec,
    const float* __restrict__ Wb2, const float* __restrict__ bb2,
    float* __restrict__ bij, unsigned* __restrict__ bmaxk)
{
  __shared__ float WcS[NBASIS * SDIM];
  __shared__ float bvS[SDIM];
  __shared__ float w2S[SDIM];
  for (int i = threadIdx.x; i < NBASIS * SDIM; i += 256) WcS[i] = Wc[i];
  for (int i = threadIdx.x; i < SDIM; i += 256) { bvS[i] = bvec[i]; w2S[i] = Wb2[i]; }
  __syncthreads();

  int lane = threadIdx.x & 31;
  int e = blockIdx.x * 8 + (threadIdx.x >> 5);
  if (e >= NEDGES) return;
  int src = ei[e];
  int dst = ei[NEDGES + e];

  float dx = pos[dst * 3 + 0] - pos[src * 3 + 0];
  float dy = pos[dst * 3 + 1] - pos[src * 3 + 1];
  float dz = pos[dst * 3 + 2] - pos[src * 3 + 2];
  float d = sqrtf(dx * dx + dy * dy + dz * dz);

  // RBF basis j = lane
  const float start = 0.006737946999085467f;              // exp(-5)
  const float g = (2.0f / (float)NBASIS) * (1.0f - start);
  const float beta = 1.0f / (g * g);
  float mean = start + (1.0f - start) * (float)lane * (1.0f / 31.0f);
  float ccut = (d < CUTR) ? 0.5f * (cosf(3.14159265358979323846f * d / CUTR) + 1.0f) : 0.0f;
  float t = expf(-d) - mean;
  float em = ccut * expf(-beta * t * t);

  const float* ub = U + (size_t)src * SDIM;
  const float* vb = V + (size_t)dst * SDIM;
  float h[8];
#pragma unroll
  for (int i = 0; i < 8; ++i) {
    int c = lane + 32 * i;
    h[i] = ub[c] + vb[c] + bvS[c];
  }
  for (int j = 0; j < NBASIS; ++j) {
    float ej = __shfl(em, j, 32);
#pragma unroll
    for (int i = 0; i < 8; ++i) h[i] += ej * WcS[j * SDIM + lane + 32 * i];
  }
  float s = 0.f;
#pragma unroll
  for (int i = 0; i < 8; ++i) {
    float x = siluf(h[i]);
    s += x * w2S[lane + 32 * i];
  }
#pragma unroll
  for (int m = 16; m >= 1; m >>= 1) s += __shfl_xor(s, m, 32);
  if (lane == 0) {
    float b = s + bb2[0];
    bij[e] = b;
    atomicMax(&bmaxk[dst], f2ord(b));
  }
}

// Pass E2: bij_exp = exp(bij - bmax[dst]); bsum[dst] += bij_exp (bsum starts at ci)
__global__ void edge_exp_kernel(const float* __restrict__ bij,
                                const int* __restrict__ ei,
                                const unsigned* __restrict__ bmaxk,
                                float* __restrict__ bije,
                                float* __restrict__ bsum)
{
  int e = blockIdx.x * blockDim.x + threadIdx.x;
  if (e >= NEDGES) return;
  int dst = ei[NEDGES + e];
  float ex = expf(bij[e] - ord2f(bmaxk[dst]));
  bije[e] = ex;
  atomicAdd(&bsum[dst], ex);
}

// Pass E3: gamma, sum_gamma, and -bij' * rhat rhat^T scattered into sigma (upper tri)
__global__ void edge_accum_kernel(const float* __restrict__ bije,
                                  const int* __restrict__ ei,
                                  const float* __restrict__ bsum,
                                  const float* __restrict__ pos,
                                  float* __restrict__ out)
{
  int e = blockIdx.x * blockDim.x + threadIdx.x;
  if (e >= NEDGES) return;
  int src = ei[e];
  int dst = ei[NEDGES + e];
  float gam = bije[e] / (bsum[dst] + 1e-7f);
  atomicAdd(&out[10 * NNODES + dst], gam);               // sum_gamma
  float bp = gam * out[9 * NNODES + dst];                // gamma * ai[dst]

  float dx = pos[dst * 3 + 0] - pos[src * 3 + 0];
  float dy = pos[dst * 3 + 1] - pos[src * 3 + 1];
  float dz = pos[dst * 3 + 2] - pos[src * 3 + 2];
  float d = sqrtf(dx * dx + dy * dy + dz * dz);
  float inv = 1.0f / (d + 1e-8f);
  float rx = dx * inv, ry = dy * inv, rz = dz * inv;

  float* sg = out + (size_t)dst * 9;
  atomicAdd(&sg[0], -bp * rx * rx);
  atomicAdd(&sg[1], -bp * rx * ry);
  atomicAdd(&sg[2], -bp * rx * rz);
  atomicAdd(&sg[4], -bp * ry * ry);
  atomicAdd(&sg[5], -bp * ry * rz);
  atomicAdd(&sg[8], -bp * rz * rz);
}

// mirror symmetric off-diagonals, add (ai + 1e-6) to diagonal
__global__ void finalize_kernel(float* __restrict__ out)
{
  int n = blockIdx.x * blockDim.x + threadIdx.x;
  if (n >= NNODES) return;
  float ain = out[9 * NNODES + n] + 1e-6f;
  float* sg = out + (size_t)n * 9;
  sg[3] = sg[1];
  sg[6] = sg[2];
  sg[7] = sg[5];
  sg[0] += ain;
  sg[4] += ain;
  sg[8] += ain;
}

extern "C" void kernel_launch(void* const* d_in, const int* in_sizes, int n_in,
                              void* d_out, int out_size, void* d_ws, size_t ws_size,
                              hipStream_t stream)
{
  const float* node_feature = (const float*)d_in[0];
  const float* pos   = (const float*)d_in[1];
  const int*   ei    = (const int*)d_in[2];
  const float* W_ai1 = (const float*)d_in[3];
  const float* b_ai1 = (const float*)d_in[4];
  const float* W_ai2 = (const float*)d_in[5];
  const float* b_ai2 = (const float*)d_in[6];
  const float* W_ci1 = (const float*)d_in[7];
  const float* b_ci1 = (const float*)d_in[8];
  const float* W_ci2 = (const float*)d_in[9];
  const float* b_ci2 = (const float*)d_in[10];
  const float* W_sc  = (const float*)d_in[11];
  const float* b_sc  = (const float*)d_in[12];
  const float* W_ed  = (const float*)d_in[13];
  const float* b_ed  = (const float*)d_in[14];
  const float* W_b1  = (const float*)d_in[15];
  const float* b_b1  = (const float*)d_in[16];
  const float* W_b2  = (const float*)d_in[17];
  const float* b_b2  = (const float*)d_in[18];
  float* out = (float*)d_out;

  // workspace layout (floats): ~64 MB total
  float* ws   = (float*)d_ws;
  float* buf0 = ws;                                  // N*256 : Hai, later U
  float* buf1 = buf0 + (size_t)NNODES * SDIM;        // N*256 : Hci, later V
  float* nsc  = buf1 + (size_t)NNODES * SDIM;        // N*256 : node_sc
  float* Wc   = nsc + (size_t)NNODES * SDIM;         // 32*256 fused rbf weight
  float* bvec = Wc + NBASIS * SDIM;                  // 256
  float* bij  = bvec + SDIM;                         // E
  float* bije = bij + NEDGES;                        // E
  unsigned* bmaxk = (unsigned*)(bije + NEDGES);      // N (ordered-key max)
  float* bsum = (float*)(bmaxk + NNODES);            // N (init = ci)

  dim3 blk(256);
  init_kernel<<<(9 * NNODES + 255) / 256, blk, 0, stream>>>(out, bmaxk);

  dim3 gN((NNODES + BM - 1) / BM, SDIM / BN);
  // stage 1: scal (first 256 cols of node_feature, row stride 960)
  gemm256<<<gN, blk, 0, stream>>>(node_feature, FDIM, NNODES, W_ai1, SDIM, b_ai1, buf0, SDIM, 1);
  gemm256<<<gN, blk, 0, stream>>>(node_feature, FDIM, NNODES, W_ci1, SDIM, b_ci1, buf1, SDIM, 1);
  gemm256<<<gN, blk, 0, stream>>>(node_feature, FDIM, NNODES, W_sc,  SDIM, b_sc,  nsc,  SDIM, 0);
  aici_kernel<<<(NNODES * 32 + 255) / 256, blk, 0, stream>>>(
      buf0, buf1, W_ai2, b_ai2, W_ci2, b_ci2, out + 9 * NNODES, bsum);
  // stage 2: U = nsc @ W_b1[0:256], V = nsc @ W_b1[256:512] (reuse buf0/buf1)
  gemm256<<<gN, blk, 0, stream>>>(nsc, SDIM, NNODES, W_b1, SDIM, nullptr, buf0, SDIM, 0);
  gemm256<<<gN, blk, 0, stream>>>(nsc, SDIM, NNODES, W_b1 + SDIM * SDIM, SDIM, nullptr, buf1, SDIM, 0);
  // fused rbf weight: Wc = W_ed(32x256) @ W_b1[512:768]
  dim3 g1(1, SDIM / BN);
  gemm256<<<g1, blk, 0, stream>>>(W_ed, SDIM, NBASIS, W_b1 + 2 * SDIM * SDIM, SDIM, nullptr, Wc, SDIM, 0);
  bvec_kernel<<<1, 256, 0, stream>>>(b_ed, W_b1, b_b1, bvec);

  edge_bij_kernel<<<(NEDGES + 7) / 8, blk, 0, stream>>>(
      buf0, buf1, ei, pos, Wc, bvec, W_b2, b_b2, bij, bmaxk);
  edge_exp_kernel<<<(NEDGES + 255) / 256, blk, 0, stream>>>(bij, ei, bmaxk, bije, bsum);
  edge_accum_kernel<<<(NEDGES + 255) / 256, blk, 0, stream>>>(bije, ei, bsum, pos, out);
  finalize_kernel<<<(NNODES + 255) / 256, blk, 0, stream>>>(out);
}